// WARPLoss_82841329205817
// MI455X (gfx1250) — compile-verified
//
#include <hip/hip_runtime.h>

#define NB 128
#define NL 1024
#define NT 64
#define ALPHA_F 1.0f
#define NEG_SENTINEL -1.0e30f

typedef float v2f __attribute__((ext_vector_type(2)));
typedef float v8f __attribute__((ext_vector_type(8)));

// ---------------------------------------------------------------------------
// Kernel 1: per-row Lmat computation.
//   Lmat[b,l] = rank_weights[(L-1)/(rej+1)] if (pos & found & valid) else 0
// ---------------------------------------------------------------------------
__global__ void lmat_kernel(const float* __restrict__ scores,
                            const float* __restrict__ target,
                            const int*   __restrict__ samples,
                            float*       __restrict__ Lmat) {
    const int b   = blockIdx.x;
    const int tid = threadIdx.x;            // 256 threads
    __shared__ float samp[NT];
    __shared__ float ranktab[NT];
    __shared__ int   s_npos;
    if (tid == 0) s_npos = 0;
    __syncthreads();

    // count positives in this row
    int local_pos = 0;
    for (int l = tid; l < NL; l += 256)
        local_pos += (target[b * NL + l] > 0.0f) ? 1 : 0;
    if (local_pos) atomicAdd(&s_npos, local_pos);

    // sampled scores + harmonic rank-weight table (computed once per block)
    if (tid < NT) {
        int si = samples[b * NT + tid];
        samp[tid] = scores[b * NL + si];
        // rank_weights[k] = H_{k+1}; we need H_{ (NL-1)/(rej+1) + 1 } for rej = tid
        int m = (NL - 1) / (tid + 1) + 1;
        float h = 0.0f;
        for (int i = 1; i <= m; ++i) h += 1.0f / (float)i;   // ascending, like cumsum
        ranktab[tid] = h;
    }
    __syncthreads();

    const int  npos  = s_npos;
    const bool valid = (npos > 0) && (npos < NL);            // num_neg = NL - npos

    for (int l = tid; l < NL; l += 256) {
        float sv  = scores[b * NL + l];
        float out = 0.0f;
        if (valid && target[b * NL + l] > 0.0f) {
            float thr = sv - ALPHA_F;                        // viol: samp > s - alpha
            int rej = -1;
            for (int t = 0; t < NT; ++t) {
                if (samp[t] > thr) { rej = t; break; }       // FIRST violating trial
            }
            if (rej >= 0) out = ranktab[rej];
        }
        Lmat[b * NL + l] = out;
    }
}

// ---------------------------------------------------------------------------
// Kernel 2: rank-weighted hinge sum via V_WMMA_F32_16X16X4_F32.
// One block (8 waves) = 8 consecutive 16-p tiles of ONE batch row.
// Row's masked scores staged once into LDS; mask folded into a -1e30 sentinel
// so the inner loop is branch-free (no EXEC churn around the WMMA).
// A[16x4] = relu(q[n] - (s_p - alpha)), B[4x16] = ones -> D rows accumulate
// per-positive hinge sums.
// ---------------------------------------------------------------------------
__global__ void loss_wmma_kernel(const float* __restrict__ scores,
                                 const float* __restrict__ target,
                                 const float* __restrict__ Lmat,
                                 float*       __restrict__ partials) {
    const int tid   = threadIdx.x;
    const int lane  = tid & 31;
    const int waveq = tid >> 5;                              // 8 waves / block
    const int tile  = blockIdx.x * 8 + waveq;                // 0 .. 8191
    const int b     = blockIdx.x >> 3;                       // 8 tiles/block, 64 tiles/row
    const int p0    = (tile & 63) << 4;                      // * 16
    const int prow  = p0 + (lane & 15);

    // Stage the row's negative-masked scores once per block.
    __shared__ float q[NL];
    {
        const float* __restrict__ srow = scores + b * NL;
        const float* __restrict__ trow = target + b * NL;
        for (int n = tid; n < NL; n += 256)
            q[n] = (trow[n] > 0.0f) ? NEG_SENTINEL : srow[n];
    }
    __syncthreads();

    const float lw = Lmat[b * NL + prow];

    // every (lane&15)==0 lane owns a partials slot; prefill so skipped tiles are 0
    const int pslot = tile * 2 + (lane >> 4);
    if ((lane & 15) == 0) partials[pslot] = 0.0f;

    // wave-uniform tile skip (keeps EXEC all-1s on the WMMA path)
    if (__ballot(lw != 0.0f) == 0ull) return;

    // hinge = relu(alpha + s_n - s_p) = relu(q[n] - (s_p - alpha))
    const float spm  = scores[b * NL + prow] - ALPHA_F;
    const int   kbase = (lane >= 16) ? 2 : 0;                // ISA A-layout: 16x4 f32

    v8f c = {};                                              // f32 accumulator tile
    v2f bones; bones[0] = 1.0f; bones[1] = 1.0f;             // B = all ones (layout-invariant)

    for (int n = 0; n < NL; n += 4) {
        const float q0 = q[n + kbase];                       // adjacent pair -> ds_load_b64
        const float q1 = q[n + kbase + 1];
        v2f a;
        a[0] = fmaxf(0.0f, q0 - spm);                        // VGPR0 -> K=kbase
        a[1] = fmaxf(0.0f, q1 - spm);                        // VGPR1 -> K=kbase+1
        c = __builtin_amdgcn_wmma_f32_16x16x4_f32(
                /*neg_a=*/false, a, /*neg_b=*/false, bones,
                /*c_mod=*/(short)0, c, /*reuse_a=*/false, /*reuse_b=*/false);
    }

    // D[i,j] = S(p0+i) for every j; read column N=0: lane 0 -> M=0..7, lane 16 -> M=8..15
    if ((lane & 15) == 0) {
        const int mbase = (lane >= 16) ? 8 : 0;
        float acc = 0.0f;
        for (int v = 0; v < 8; ++v)
            acc += Lmat[b * NL + p0 + mbase + v] * c[v];
        partials[pslot] = acc;
    }
}

// ---------------------------------------------------------------------------
// Kernel 3: deterministic fixed-order reduction of 16384 partials -> loss.
// ---------------------------------------------------------------------------
__global__ void reduce_kernel(const float* __restrict__ partials,
                              float*       __restrict__ out) {
    __shared__ float sm[256];
    const int tid = threadIdx.x;
    float s = 0.0f;
    for (int i = tid; i < NB * (NL / 16) * 2; i += 256) s += partials[i];
    sm[tid] = s;
    __syncthreads();
    for (int w = 128; w > 0; w >>= 1) {
        if (tid < w) sm[tid] += sm[tid + w];
        __syncthreads();
    }
    if (tid == 0) out[0] = sm[0];
}

// ---------------------------------------------------------------------------
extern "C" void kernel_launch(void* const* d_in, const int* in_sizes, int n_in,
                              void* d_out, int out_size, void* d_ws, size_t ws_size,
                              hipStream_t stream) {
    const float* scores  = (const float*)d_in[0];
    const float* target  = (const float*)d_in[1];
    const int*   samples = (const int*)d_in[2];
    float*       out     = (float*)d_out;

    float* lmat     = (float*)d_ws;                              // NB*NL floats (512 KB)
    float* partials = (float*)((char*)d_ws + (size_t)NB * NL * sizeof(float)); // 16384 floats

    lmat_kernel<<<NB, 256, 0, stream>>>(scores, target, samples, lmat);
    loss_wmma_kernel<<<(NB * (NL / 16)) / 8, 256, 0, stream>>>(scores, target, lmat, partials);
    reduce_kernel<<<1, 256, 0, stream>>>(partials, out);
}